// LocalGlobalAttention_71244917506118
// MI455X (gfx1250) — compile-verified
//
#include <hip/hip_runtime.h>
#include <hip/hip_bf16.h>
#include <math.h>
#include <stdint.h>

#define SEQ     2048
#define DIM     512
#define HEADS   8
#define HEAD_DIM 64
#define WINDOW  128
#define NEGBIAS (-10000.0f)

typedef float v2f __attribute__((ext_vector_type(2)));
typedef float v8f __attribute__((ext_vector_type(8)));

// D = A(16x4,f32) * B(4x16,f32) + C(16x16,f32)   -> v_wmma_f32_16x16x4_f32
static __device__ __forceinline__ v8f wmma4(v2f a, v2f b, v8f c) {
    return __builtin_amdgcn_wmma_f32_16x16x4_f32(
        /*neg_a=*/false, a, /*neg_b=*/false, b,
        /*c_mod=*/(short)0, c, /*reuse_a=*/false, /*reuse_b=*/false);
}
static __device__ __forceinline__ v2f ldv2(const float* p) {
    return *reinterpret_cast<const v2f*>(p);
}

// CDNA5 async global->LDS copy (ASYNCcnt path). VDST = LDS byte address VGPR,
// ADDR = 64-bit global address VGPR pair. 16 bytes per lane per issue.
static __device__ __forceinline__ void async_load_b128(uint32_t lds_off, const float* gptr) {
    asm volatile("global_load_async_to_lds_b128 %0, %1, off"
                 :: "v"(lds_off), "v"(gptr)
                 : "memory");
}
static __device__ __forceinline__ void wait_asynccnt0() {
    asm volatile("s_wait_asynccnt 0" ::: "memory");
}

// ---------------------------------------------------------------------------
// Kernel 1: out[2048x512] = X[2048x512] @ W^T[512x512] + bias, then * scale.
// Block = 256 threads (8 waves) covers 16 rows x 512 cols; wave w owns 64 cols.
// f32 WMMA layout: A-frag lane(l<16): M=l, k={0,1}; lane(l>=16): M=l-16, k={2,3}.
// B-frag mirrored on the K dimension; C: vgpr r -> row r (+8 for upper half).
// ---------------------------------------------------------------------------
__global__ __launch_bounds__(256) void proj_kernel(
    const float* __restrict__ X, const float* __restrict__ W,
    const float* __restrict__ bias, float* __restrict__ out, float scale)
{
    const int tid  = threadIdx.x;
    const int wv   = tid >> 5;
    const int lane = tid & 31;
    const int half = lane >> 4;
    const int m    = lane & 15;
    const int q0   = blockIdx.x * 16;
    const int nb   = wv * 64;

    v8f c[4] = {};
    const float* arow = X + (size_t)(q0 + m) * DIM + 2 * half;
    #pragma unroll 4
    for (int k = 0; k < DIM / 4; ++k) {
        v2f a = ldv2(arow + 4 * k);
        #pragma unroll
        for (int nt = 0; nt < 4; ++nt) {
            v2f b = ldv2(W + (size_t)(nb + nt * 16 + m) * DIM + 4 * k + 2 * half);
            c[nt] = wmma4(a, b, c[nt]);
        }
    }
    #pragma unroll
    for (int nt = 0; nt < 4; ++nt) {
        const int col = nb + nt * 16 + m;
        const float bn = bias[col];
        #pragma unroll
        for (int r = 0; r < 8; ++r) {
            const int row = q0 + r + 8 * half;
            out[(size_t)row * DIM + col] = (c[nt][r] + bn) * scale;
        }
    }
}

// ---------------------------------------------------------------------------
// Kernel 2: banded local attention, one block per (16-query-row tile, head).
// 8 waves; wave w owns key columns [w*256, w*256+256) as 16 WMMA score tiles
// kept entirely in VGPRs (16 x v8f = 128 regs). Softmax: shuffle-reduce within
// wave halves, then tiny LDS cross-wave reduce. Normalized P goes straight to
// the dense attn output and (via a wave-private LDS tile) into the PV WMMAs.
// V tiles are staged to LDS with global_load_async_to_lds_b128 (coalesced,
// ASYNCcnt-tracked) instead of strided scalar loads.
// ---------------------------------------------------------------------------
__global__ __launch_bounds__(256) void local_attn_kernel(
    const float* __restrict__ Q, const float* __restrict__ K,
    const float* __restrict__ V, const int* __restrict__ mask,
    float* __restrict__ out, float* __restrict__ attn)
{
    __shared__ float lds_p[8][16][16];     // per-wave P tile staging (A-layout feed)
    __shared__ float redmax[8][16];
    __shared__ float redsum[8][16];
    // Phase-disjoint alias: V-tile stage during the PV loop (wave-private,
    // async writes fenced by s_wait_asynccnt), ctx partials afterwards.
    __shared__ float smem_vc[8][16][64];

    const int tid  = threadIdx.x;
    const int wv   = tid >> 5;
    const int lane = tid & 31;
    const int half = lane >> 4;
    const int m    = lane & 15;
    const int qt   = blockIdx.x;
    const int h    = blockIdx.y;
    const int q0   = qt * 16;
    const size_t hb = (size_t)h * HEAD_DIM;

    // query-row mask flags (C-layout rows owned by this lane)
    int irow[8]; int gi[8]; int pi[8];
    #pragma unroll
    for (int r = 0; r < 8; ++r) {
        const int i = q0 + r + 8 * half;
        irow[r] = i;
        const int mi = mask[i];
        gi[r] = (mi == -1);
        pi[r] = (mi == 0);
    }

    // preload all A-fragments of the 16x64 Q tile (Q is pre-scaled by 1/8)
    v2f af[16];
    const float* qrow = Q + (size_t)(q0 + m) * DIM + hb + 2 * half;
    #pragma unroll
    for (int k = 0; k < 16; ++k) af[k] = ldv2(qrow + 4 * k);

    // ---- scores = Q K^T + bias, held in registers ----
    v8f sacc[16];
    #pragma unroll
    for (int kt = 0; kt < 16; ++kt) {
        const int j0 = wv * 256 + kt * 16;
        if (kt < 15)
            __builtin_prefetch(K + (size_t)(j0 + 16 + m) * DIM + hb, 0, 1);
        v8f c = {};
        const float* krow = K + (size_t)(j0 + m) * DIM + hb + 2 * half;
        #pragma unroll
        for (int k = 0; k < 16; ++k) {
            v2f b = ldv2(krow + 4 * k);
            c = wmma4(af[k], b, c);
        }
        const int j  = j0 + m;
        const int mj = mask[j];
        const int cg = (mj == -1);
        const int cp = (mj == 0);
        #pragma unroll
        for (int r = 0; r < 8; ++r) {
            const int i = irow[r];
            const int d = (i > j) ? (i - j) : (j - i);
            const bool allowed = ((d <= WINDOW) || cg || gi[r]) && !cp && !pi[r];
            if (!allowed) c[r] += NEGBIAS;
        }
        sacc[kt] = c;
    }

    // ---- row max (wave-local shuffle reduce, then cross-wave via LDS) ----
    float lmax[8];
    #pragma unroll
    for (int r = 0; r < 8; ++r) lmax[r] = -3.0e38f;
    #pragma unroll
    for (int kt = 0; kt < 16; ++kt)
        #pragma unroll
        for (int r = 0; r < 8; ++r) lmax[r] = fmaxf(lmax[r], sacc[kt][r]);
    #pragma unroll
    for (int xm = 1; xm <= 8; xm <<= 1)
        #pragma unroll
        for (int r = 0; r < 8; ++r)
            lmax[r] = fmaxf(lmax[r], __shfl_xor(lmax[r], xm, 32));
    if (m == 0) {
        #pragma unroll
        for (int r = 0; r < 8; ++r) redmax[wv][8 * half + r] = lmax[r];
    }
    __syncthreads();
    float rmax[8];
    #pragma unroll
    for (int r = 0; r < 8; ++r) {
        float v = -3.0e38f;
        #pragma unroll
        for (int w = 0; w < 8; ++w) v = fmaxf(v, redmax[w][8 * half + r]);
        rmax[r] = v;
    }

    // ---- exp + row sum ----
    float lsum[8] = {};
    #pragma unroll
    for (int kt = 0; kt < 16; ++kt)
        #pragma unroll
        for (int r = 0; r < 8; ++r) {
            const float p = __expf(sacc[kt][r] - rmax[r]);
            sacc[kt][r] = p;
            lsum[r] += p;
        }
    #pragma unroll
    for (int xm = 1; xm <= 8; xm <<= 1)
        #pragma unroll
        for (int r = 0; r < 8; ++r) lsum[r] += __shfl_xor(lsum[r], xm, 32);
    if (m == 0) {
        #pragma unroll
        for (int r = 0; r < 8; ++r) redsum[wv][8 * half + r] = lsum[r];
    }
    __syncthreads();
    float inv[8];
    #pragma unroll
    for (int r = 0; r < 8; ++r) {
        float s = 0.f;
        #pragma unroll
        for (int w = 0; w < 8; ++w) s += redsum[w][8 * half + r];
        const float v = 1.0f / s;
        inv[r] = pi[r] ? 0.0f : v;   // zero padded query rows (reference rule)
    }

    // ---- normalize, store dense attn probabilities ----
    float* attn_h = attn + (size_t)h * SEQ * SEQ;
    #pragma unroll
    for (int kt = 0; kt < 16; ++kt) {
        const int j0 = wv * 256 + kt * 16;
        #pragma unroll
        for (int r = 0; r < 8; ++r) {
            const float p = sacc[kt][r] * inv[r];
            sacc[kt][r] = p;
            attn_h[(size_t)irow[r] * SEQ + j0 + m] = p;
        }
    }

    // ---- ctx = P @ V via WMMA; V tile async-staged to LDS per key tile ----
    const uint32_t vbase = (uint32_t)(uintptr_t)&smem_vc[wv][0][0];
    v8f cacc[4] = {};
    #pragma unroll
    for (int kt = 0; kt < 16; ++kt) {
        const int jb = wv * 256 + kt * 16;

        // async-copy this wave's 16x64 fp32 V tile (4 KB) into LDS:
        // 8 issues x 32 lanes x 16 B, fully coalesced.
        #pragma unroll
        for (int it = 0; it < 8; ++it) {
            const int c   = it * 32 + lane;
            const int row = c >> 4;
            const int q4  = (c & 15) * 4;
            async_load_b128(vbase + (uint32_t)(row * 64 + q4) * 4u,
                            V + (size_t)(jb + row) * DIM + hb + q4);
        }

        // P tile to wave-private LDS for the A-layout feed (independent of async)
        #pragma unroll
        for (int r = 0; r < 8; ++r) lds_p[wv][r + 8 * half][m] = sacc[kt][r];

        wait_asynccnt0();   // V tile resident in LDS

        #pragma unroll
        for (int k4 = 0; k4 < 4; ++k4) {
            v2f a = *reinterpret_cast<const v2f*>(&lds_p[wv][m][4 * k4 + 2 * half]);
            const int kkl = 4 * k4 + 2 * half;
            #pragma unroll
            for (int nt = 0; nt < 4; ++nt) {
                v2f b;
                b[0] = smem_vc[wv][kkl][nt * 16 + m];
                b[1] = smem_vc[wv][kkl + 1][nt * 16 + m];
                cacc[nt] = wmma4(a, b, cacc[nt]);
            }
        }
        // Next iteration's async stores can't overtake these LDS reads: the
        // WMMA consumers force the compiler's s_wait_dscnt before reuse.
    }

    // ---- cross-wave ctx reduction (reusing smem_vc as partial buffer) ----
    #pragma unroll
    for (int nt = 0; nt < 4; ++nt)
        #pragma unroll
        for (int r = 0; r < 8; ++r)
            smem_vc[wv][r + 8 * half][nt * 16 + m] = cacc[nt][r];
    __syncthreads();
    for (int e = tid; e < 16 * 64; e += 256) {
        const int rr = e >> 6, dd = e & 63;
        float s = 0.f;
        #pragma unroll
        for (int w = 0; w < 8; ++w) s += smem_vc[w][rr][dd];
        out[(size_t)(q0 + rr) * DIM + hb + dd] = s;
    }
}

// ---------------------------------------------------------------------------
// Kernel 3: global-token attention. Only rows with mask==-1 are active
// (~16 rows x 8 heads => tiny). Overwrites those rows of `out` (stream order
// after kernel 2 implements where(is_glob, gctx, ctx)).
// ---------------------------------------------------------------------------
__global__ __launch_bounds__(256) void global_attn_kernel(
    const float* __restrict__ Qg, const float* __restrict__ Kg,
    const float* __restrict__ Vg, const int* __restrict__ mask,
    float* __restrict__ out)
{
    const int row = blockIdx.x;
    const int h   = blockIdx.y;
    if (mask[row] != -1) return;          // uniform per block

    const int tid = threadIdx.x;
    const size_t hb = (size_t)h * HEAD_DIM;
    __shared__ float red[256];
    __shared__ float sctx[64];

    float qv[64];
    #pragma unroll
    for (int d = 0; d < 64; ++d) qv[d] = Qg[(size_t)row * DIM + hb + d];

    float lmax = -3.0e38f;
    for (int j = tid; j < SEQ; j += 256) {
        const float* kr = Kg + (size_t)j * DIM + hb;
        float s = 0.f;
        #pragma unroll
        for (int d = 0; d < 64; ++d) s += qv[d] * kr[d];
        if (mask[j] == 0) s = NEGBIAS;
        lmax = fmaxf(lmax, s);
    }
    red[tid] = lmax; __syncthreads();
    for (int st = 128; st > 0; st >>= 1) {
        if (tid < st) red[tid] = fmaxf(red[tid], red[tid + st]);
        __syncthreads();
    }
    const float smax = red[0];
    __syncthreads();

    float lsum = 0.f;
    float acc[64] = {};
    for (int j = tid; j < SEQ; j += 256) {
        const float* kr = Kg + (size_t)j * DIM + hb;
        float s = 0.f;
        #pragma unroll
        for (int d = 0; d < 64; ++d) s += qv[d] * kr[d];
        if (mask[j] == 0) s = NEGBIAS;
        const float p = __expf(s - smax);
        lsum += p;
        const float* vr = Vg + (size_t)j * DIM + hb;
        #pragma unroll
        for (int d = 0; d < 64; ++d) acc[d] += p * vr[d];
    }
    red[tid] = lsum; __syncthreads();
    for (int st = 128; st > 0; st >>= 1) {
        if (tid < st) red[tid] += red[tid + st];
        __syncthreads();
    }
    const float ssum = red[0];
    if (tid < 64) sctx[tid] = 0.f;
    __syncthreads();
    #pragma unroll
    for (int d = 0; d < 64; ++d) atomicAdd(&sctx[d], acc[d]);   // ds_add_f32
    __syncthreads();
    if (tid < 64) out[(size_t)row * DIM + hb + tid] = sctx[tid] / ssum;
}

// ---------------------------------------------------------------------------
// Host side. Input order: X, Wq,bq, Wk,bk, Wv,bv, Wqg,bqg, Wkg,bkg, Wvg,bvg, mask.
// d_out = out (SEQ*DIM floats) ++ attn (HEADS*SEQ*SEQ floats).
// Workspace: 6 projection matrices of SEQ*DIM fp32 = 24 MiB.
// ---------------------------------------------------------------------------
extern "C" void kernel_launch(void* const* d_in, const int* in_sizes, int n_in,
                              void* d_out, int out_size, void* d_ws, size_t ws_size,
                              hipStream_t stream) {
    (void)in_sizes; (void)n_in; (void)out_size; (void)ws_size;

    const float* X    = (const float*)d_in[0];
    const int*   mask = (const int*)d_in[13];
    float* out  = (float*)d_out;
    float* attn = out + (size_t)SEQ * DIM;

    const size_t SD = (size_t)SEQ * DIM;
    float* ws = (float*)d_ws;
    float* bufs[6];
    for (int i = 0; i < 6; ++i) bufs[i] = ws + (size_t)i * SD;

    const float scale = 0.125f;  // 1/sqrt(HEAD_DIM)
    for (int mt = 0; mt < 6; ++mt) {
        const float* W = (const float*)d_in[1 + 2 * mt];
        const float* b = (const float*)d_in[2 + 2 * mt];
        const float sc = (mt == 0 || mt == 3) ? scale : 1.0f;
        proj_kernel<<<dim3(SEQ / 16), 256, 0, stream>>>(X, W, b, bufs[mt], sc);
    }
    local_attn_kernel<<<dim3(SEQ / 16, HEADS), 256, 0, stream>>>(
        bufs[0], bufs[1], bufs[2], mask, out, attn);
    global_attn_kernel<<<dim3(SEQ, HEADS), 256, 0, stream>>>(
        bufs[3], bufs[4], bufs[5], mask, out);
}